// HGLP_25451976196825
// MI455X (gfx1250) — compile-verified
//
#include <hip/hip_runtime.h>
#include <hip/hip_bf16.h>

#define N_NODES  50000
#define N_HEDGES 50000
#define N_INC    800000
#define N_LINKS  20000

typedef __attribute__((ext_vector_type(2))) float v2f;
typedef __attribute__((ext_vector_type(8))) float v8f;

__device__ __forceinline__ void atomAddF(float* p, float v) {
    // native global_atomic_add_f32 (no CAS loop)
    unsafeAtomicAdd(p, v);
}

// ---------------------------------------------------------------- utility
__global__ void k_zero(float* __restrict__ p, size_t n) {
    size_t t = (size_t)blockIdx.x * blockDim.x + threadIdx.x;
    if (t < n) p[t] = 0.f;
}

__global__ void k_deg(const int* __restrict__ src, const int* __restrict__ dst,
                      float* __restrict__ dn, float* __restrict__ dh, int n) {
    int t = blockIdx.x * blockDim.x + threadIdx.x;
    if (t < n) {
        atomAddF(dn + src[t], 1.f);
        atomAddF(dh + dst[t], 1.f);
    }
}

__global__ void k_inv(float* __restrict__ p, int n) {
    int t = blockIdx.x * blockDim.x + threadIdx.x;
    if (t < n) { float v = p[t]; p[t] = v > 0.f ? 1.f / v : 0.f; }
}

// ------------------------------------------------- WMMA f32 GEMM (exact)
// C[M,N] = A[M,K] @ B[K,N] (+bias, optional ReLU). One wave per 16x16 tile.
// Uses V_WMMA_F32_16X16X4_F32: A frag = 2 VGPRs (lane=row M, half selects K
// pair), B frag = 2 VGPRs (lane=col N), C/D = 8 VGPRs (M = i + 8*half).
__global__ void k_gemm_wmma(const float* __restrict__ A, const float* __restrict__ B,
                            float* __restrict__ C, int M, int K, int N,
                            const float* __restrict__ bias, int relu)
{
    int wave = (blockIdx.x * blockDim.x + threadIdx.x) >> 5;
    int lane = threadIdx.x & 31;
    int ntN  = N >> 4;
    int mt   = wave / ntN;
    int nt   = wave - mt * ntN;
    if (mt >= (M >> 4)) return;                 // wave-uniform: EXEC stays all-1s
    int m0 = mt << 4, n0 = nt << 4;
    int half = lane >> 4, r = lane & 15;

    const float* ap = A + (size_t)(m0 + r) * K + 2 * half;  // 8B-aligned pair
    const float* bp = B + (size_t)(2 * half) * N + n0 + r;

    v8f acc = {};
    for (int k0 = 0; k0 < K; k0 += 4) {
        v2f a = *(const v2f*)(ap + k0);         // A[m0+r, k0+2h .. +1]
        v2f b;
        b.x = bp[(size_t)k0 * N];               // B[k0+2h,   n0+r]
        b.y = bp[(size_t)(k0 + 1) * N];         // B[k0+2h+1, n0+r]
        acc = __builtin_amdgcn_wmma_f32_16x16x4_f32(
            false, a, false, b, (short)0, acc, false, false);
    }

    float bv = bias ? bias[n0 + r] : 0.f;
#pragma unroll
    for (int i = 0; i < 8; ++i) {
        float v = acc[i] + bv;
        if (relu) v = fmaxf(v, 0.f);
        C[(size_t)(m0 + i + 8 * half) * N + n0 + r] = v;
    }
}

// ------------------------------------------------------- sparse scatters
// he[dst[i], :] += xw[src[i], :]
__global__ void k_scat1(const int* __restrict__ src, const int* __restrict__ dst,
                        const float* __restrict__ xw, float* __restrict__ he,
                        unsigned d4, int d, unsigned total)
{
    unsigned t = blockIdx.x * blockDim.x + threadIdx.x;
    if (t >= total) return;
    unsigned i = t / d4;
    unsigned c = (t - i * d4) * 4u;
    const float4 v = *(const float4*)(xw + (size_t)src[i] * d + c);
    float* p = he + (size_t)dst[i] * d + c;
    atomAddF(p + 0, v.x); atomAddF(p + 1, v.y);
    atomAddF(p + 2, v.z); atomAddF(p + 3, v.w);
}

// out[src[i], :] += he[dst[i], :] * binv[dst[i]]   (B^{-1} folded in)
__global__ void k_scat2(const int* __restrict__ src, const int* __restrict__ dst,
                        const float* __restrict__ he, const float* __restrict__ binv,
                        float* __restrict__ out, unsigned d4, int d, unsigned total)
{
    unsigned t = blockIdx.x * blockDim.x + threadIdx.x;
    if (t >= total) return;
    unsigned i = t / d4;
    unsigned c = (t - i * d4) * 4u;
    int dd = dst[i];
    float bi = binv[dd];
    const float4 v = *(const float4*)(he + (size_t)dd * d + c);
    float* p = out + (size_t)src[i] * d + c;
    atomAddF(p + 0, v.x * bi); atomAddF(p + 1, v.y * bi);
    atomAddF(p + 2, v.z * bi); atomAddF(p + 3, v.w * bi);
}

// out = relu(out * dinv[row] + bias[col])
__global__ void k_final(float* __restrict__ out, const float* __restrict__ dinv,
                        const float* __restrict__ bias, int d, unsigned n)
{
    unsigned t = blockIdx.x * blockDim.x + threadIdx.x;
    if (t >= n) return;
    unsigned r = t / (unsigned)d, c = t - r * (unsigned)d;
    out[t] = fmaxf(out[t] * dinv[r] + bias[c], 0.f);
}

// ------------------------------------------------------ feature assembly
// feat[l] = [ min(ec[e0],ec[e1]) (192) | max(...) (192) | nxcat[marks[l]] (384) ]
__global__ void k_feat(const float* __restrict__ nx0, const float* __restrict__ nx1,
                       const float* __restrict__ nx2, const float* __restrict__ ex0,
                       const float* __restrict__ ex1, const float* __restrict__ ex2,
                       const int* __restrict__ marks, const int* __restrict__ emarks,
                       float* __restrict__ feat, unsigned total)
{
    unsigned t = blockIdx.x * blockDim.x + threadIdx.x;
    if (t >= total) return;
    unsigned l = t / 768u, c = t - l * 768u;
    float v;
    if (c < 384u) {
        unsigned cc = (c < 192u) ? c : c - 192u;
        int e0 = emarks[l], e1 = e0 + 1;
        float a, b;
        if (cc < 64u)       { a = ex0[(size_t)e0 * 64 + cc];        b = ex0[(size_t)e1 * 64 + cc]; }
        else if (cc < 128u) { a = ex1[(size_t)e0 * 64 + cc - 64];   b = ex1[(size_t)e1 * 64 + cc - 64]; }
        else                { a = ex2[(size_t)e0 * 64 + cc - 128];  b = ex2[(size_t)e1 * 64 + cc - 128]; }
        v = (c < 192u) ? fminf(a, b) : fmaxf(a, b);
    } else {
        unsigned cc = c - 384u;
        int m = marks[l];
        if (cc < 128u)      v = nx0[(size_t)m * 128 + cc];
        else if (cc < 256u) v = nx1[(size_t)m * 128 + cc - 128];
        else                v = nx2[(size_t)m * 128 + cc - 256];
    }
    feat[t] = v;
}

// ------------------------------------------- 256 -> 2 head + log_softmax
__global__ void k_head(const float* __restrict__ h1, const float* __restrict__ W2,
                       const float* __restrict__ b2, float* __restrict__ out, int n)
{
    int l = blockIdx.x * blockDim.x + threadIdx.x;
    if (l >= n) return;
    const float* h = h1 + (size_t)l * 256;
    float z0 = b2[0], z1 = b2[1];
#pragma unroll 4
    for (int k = 0; k < 256; ++k) {
        float hv = h[k];
        z0 += hv * W2[2 * k];
        z1 += hv * W2[2 * k + 1];
    }
    float m   = fmaxf(z0, z1);
    float lse = m + logf(expf(z0 - m) + expf(z1 - m));
    out[2 * l]     = z0 - lse;
    out[2 * l + 1] = z1 - lse;
}

// ---------------------------------------------------------------- host
static inline unsigned gridFor(size_t n) { return (unsigned)((n + 255) / 256); }

static void launch_gemm(const float* A, const float* B, float* C,
                        int M, int K, int N, const float* bias, int relu,
                        hipStream_t s)
{
    int waves = (M >> 4) * (N >> 4);
    int blocks = (waves + 7) / 8;               // 8 wave32 per 256-thread block
    k_gemm_wmma<<<blocks, 256, 0, s>>>(A, B, C, M, K, N, bias, relu);
}

// One HypergraphConv layer: out = relu(D^{-1} H B^{-1} H^T (h W) + b)
static void hconv(const float* h, const float* W, const float* bias,
                  const int* src, const int* dst,
                  const float* binv, const float* dinv,
                  int n_src, int n_dst, int din, int dout,
                  float* xw, float* hebuf, float* out, hipStream_t s)
{
    launch_gemm(h, W, xw, n_src, din, dout, nullptr, 0, s);

    size_t nhe = (size_t)n_dst * dout;
    k_zero<<<gridFor(nhe), 256, 0, s>>>(hebuf, nhe);
    unsigned d4 = (unsigned)(dout / 4);
    unsigned tot = (unsigned)N_INC * d4;
    k_scat1<<<gridFor(tot), 256, 0, s>>>(src, dst, xw, hebuf, d4, dout, tot);

    size_t nout = (size_t)n_src * dout;
    k_zero<<<gridFor(nout), 256, 0, s>>>(out, nout);
    k_scat2<<<gridFor(tot), 256, 0, s>>>(src, dst, hebuf, binv, out, d4, dout, tot);
    k_final<<<gridFor(nout), 256, 0, s>>>(out, dinv, bias, dout, (unsigned)nout);
}

extern "C" void kernel_launch(void* const* d_in, const int* in_sizes, int n_in,
                              void* d_out, int out_size, void* d_ws, size_t ws_size,
                              hipStream_t stream)
{
    const float* x      = (const float*)d_in[0];   // [50000, 256]
    const float* edge_x = (const float*)d_in[1];   // [50000, 128]
    const int*   einc   = (const int*)d_in[2];     // [2, 800000]
    const int*   nsrc   = einc;                    // node indices
    const int*   hdst   = einc + N_INC;            // hyperedge indices
    const int*   marks  = (const int*)d_in[3];
    const int*   emarks = (const int*)d_in[4];
    const float* Wn[3]  = {(const float*)d_in[5],  (const float*)d_in[9],  (const float*)d_in[13]};
    const float* bn[3]  = {(const float*)d_in[6],  (const float*)d_in[10], (const float*)d_in[14]};
    const float* We[3]  = {(const float*)d_in[7],  (const float*)d_in[11], (const float*)d_in[15]};
    const float* be[3]  = {(const float*)d_in[8],  (const float*)d_in[12], (const float*)d_in[16]};
    const float* W1 = (const float*)d_in[17];
    const float* b1 = (const float*)d_in[18];
    const float* W2 = (const float*)d_in[19];
    const float* b2 = (const float*)d_in[20];

    // ---- workspace carve-up (floats)
    float* ws    = (float*)d_ws;
    float* deg_n = ws;  ws += N_NODES;             // node degrees -> D^{-1}(node)
    float* deg_h = ws;  ws += N_HEDGES;            // hedge degrees -> B^{-1}(node conv)
    float* nx[3];
    for (int i = 0; i < 3; ++i) { nx[i] = ws; ws += (size_t)N_NODES * 128; }
    float* ex[3];
    for (int i = 0; i < 3; ++i) { ex[i] = ws; ws += (size_t)N_HEDGES * 64; }
    float* xw   = ws;   ws += (size_t)N_NODES * 128;
    float* he   = ws;   ws += (size_t)N_NODES * 128;
    float* feat = ws;   ws += (size_t)N_LINKS * 768;
    float* h1   = ws;   ws += (size_t)N_LINKS * 256;

    // ---- degrees (shared by all 6 conv layers), then invert in place
    k_zero<<<gridFor(N_NODES),  256, 0, stream>>>(deg_n, N_NODES);
    k_zero<<<gridFor(N_HEDGES), 256, 0, stream>>>(deg_h, N_HEDGES);
    k_deg<<<gridFor(N_INC), 256, 0, stream>>>(nsrc, hdst, deg_n, deg_h, N_INC);
    k_inv<<<gridFor(N_NODES),  256, 0, stream>>>(deg_n, N_NODES);
    k_inv<<<gridFor(N_HEDGES), 256, 0, stream>>>(deg_h, N_HEDGES);

    // ---- node convs: src=nodes, dst=hedges; B^{-1}=deg_h^{-1}, D^{-1}=deg_n^{-1}
    hconv(x,     Wn[0], bn[0], nsrc, hdst, deg_h, deg_n, N_NODES, N_HEDGES, 256, 128, xw, he, nx[0], stream);
    hconv(nx[0], Wn[1], bn[1], nsrc, hdst, deg_h, deg_n, N_NODES, N_HEDGES, 128, 128, xw, he, nx[1], stream);
    hconv(nx[1], Wn[2], bn[2], nsrc, hdst, deg_h, deg_n, N_NODES, N_HEDGES, 128, 128, xw, he, nx[2], stream);

    // ---- edge convs (flipped incidence): src=hedges, dst=nodes; inverses swap
    hconv(edge_x, We[0], be[0], hdst, nsrc, deg_n, deg_h, N_HEDGES, N_NODES, 128, 64, xw, he, ex[0], stream);
    hconv(ex[0],  We[1], be[1], hdst, nsrc, deg_n, deg_h, N_HEDGES, N_NODES,  64, 64, xw, he, ex[1], stream);
    hconv(ex[1],  We[2], be[2], hdst, nsrc, deg_n, deg_h, N_HEDGES, N_NODES,  64, 64, xw, he, ex[2], stream);

    // ---- link features, MLP (WMMA, fused bias+ReLU), head
    unsigned ftot = (unsigned)N_LINKS * 768u;
    k_feat<<<gridFor(ftot), 256, 0, stream>>>(nx[0], nx[1], nx[2], ex[0], ex[1], ex[2],
                                              marks, emarks, feat, ftot);
    launch_gemm(feat, W1, h1, N_LINKS, 768, 256, b1, 1, stream);
    k_head<<<gridFor(N_LINKS), 256, 0, stream>>>(h1, W2, b2, (float*)d_out, N_LINKS);
}